// GIN_57140244906477
// MI455X (gfx1250) — compile-verified
//
#include <hip/hip_runtime.h>

#define NNODES 100000
#define NEDGES 1600000
#define HID    64
#define NTILES (NNODES / 16)      // 6250, exact
#define BN_EPS 1e-5f
#define WSTRIDE 68                // padded transposed-weight row stride (dwords)

typedef __attribute__((ext_vector_type(2))) float v2f;
typedef __attribute__((ext_vector_type(8))) float v8f;

// ---------------------------------------------------------------------------
// agg = x  (fused "+ x_i" self term of GINConv), and zero BN stats [128 floats]
// ---------------------------------------------------------------------------
__global__ void __launch_bounds__(256)
gin_init(const float* __restrict__ xin, float* __restrict__ agg,
         float* __restrict__ stats) {
  int gid = blockIdx.x * 256 + threadIdx.x;
  if (gid < NNODES * (HID / 4)) {
    ((float4*)agg)[gid] = ((const float4*)xin)[gid];
  }
  if (gid < 2 * HID) stats[gid] = 0.0f;
}

// ---------------------------------------------------------------------------
// One wave per edge: agg[dst] += x[src] * atten.  L2-resident accumulator,
// float2 per lane -> a wave covers one full 256B feature row, coalesced.
// ---------------------------------------------------------------------------
__global__ void __launch_bounds__(256)
gin_edge(const float* __restrict__ x, const long long* __restrict__ ei,
         const float* __restrict__ atten, float* __restrict__ agg) {
  int gid  = blockIdx.x * 256 + threadIdx.x;
  int e    = gid >> 5;
  int lane = gid & 31;
  if (e >= NEDGES) return;
  long long s = ei[e];
  long long d = ei[NEDGES + e];
  float a = atten[e];
  v2f v = *(const v2f*)(x + (size_t)s * HID + lane * 2);
  float* dr = agg + (size_t)d * HID + lane * 2;
  atomicAdd(dr,     v.x * a);
  atomicAdd(dr + 1, v.y * a);
}

// ---------------------------------------------------------------------------
// Weights staged in LDS TRANSPOSED with padded stride 68:
//   sWT[n * 68 + k] = W[k * 64 + n]
// so a B fragment {W[ka][n], W[ka+1][n]} is one aligned ds_load_b64 and the
// 32 lanes of a wave hit 64 distinct banks (bank = (4n + ka) mod 64).
// ---------------------------------------------------------------------------
__device__ __forceinline__ void load_wt(float* sWT, const float* __restrict__ W,
                                        int tid) {
  for (int i = tid; i < HID * HID; i += 128) {
    int k = i >> 6;          // row of W   (K index)
    int n = i & 63;          // col of W   (N index)
    sWT[n * WSTRIDE + k] = W[i];
  }
}

// ---------------------------------------------------------------------------
// GEMM1: h1 = agg @ W1 + b1  (fp32 WMMA 16x16x4), plus per-column sum/sumsq
// for BatchNorm via atomics.  One wave per 16-row tile, all 64 columns.
// A 16x4 fp32 layout: lanes 0-15 = rows M with K{0,1} in the VGPR pair;
// lanes 16-31 carry K{2,3}.  B 4x16 mirrors with lanes carrying N.
// ---------------------------------------------------------------------------
__global__ void __launch_bounds__(128)
gin_gemm1(const float* __restrict__ agg, const float* __restrict__ W,
          const float* __restrict__ bias, float* __restrict__ h1,
          float* __restrict__ stats) {
  __shared__ float sWT[HID * WSTRIDE];
  load_wt(sWT, W, threadIdx.x);
  __syncthreads();

  int lane = threadIdx.x & 31;
  int tile = blockIdx.x * 4 + (threadIdx.x >> 5);
  if (tile >= NTILES) return;          // wave-uniform: EXEC stays all-ones

  int nl = lane & 15;                  // row index (A) / column index (B,D)
  int kh = lane >> 4;                  // K-half selector
  int m0 = tile * 16;
  const float* arow = agg + (size_t)(m0 + nl) * HID;
  const float* w0 = sWT + (size_t)nl * WSTRIDE;

  v8f acc0 = {}, acc1 = {}, acc2 = {}, acc3 = {};
  for (int k = 0; k < HID; k += 4) {
    int ka = k + 2 * kh;
    v2f a  = *(const v2f*)(arow + ka);
    v2f b0 = *(const v2f*)(w0 + ka);
    v2f b1 = *(const v2f*)(w0 + 16 * WSTRIDE + ka);
    v2f b2 = *(const v2f*)(w0 + 32 * WSTRIDE + ka);
    v2f b3 = *(const v2f*)(w0 + 48 * WSTRIDE + ka);
    acc0 = __builtin_amdgcn_wmma_f32_16x16x4_f32(false, a, false, b0, (short)0, acc0, false, false);
    acc1 = __builtin_amdgcn_wmma_f32_16x16x4_f32(false, a, false, b1, (short)0, acc1, false, false);
    acc2 = __builtin_amdgcn_wmma_f32_16x16x4_f32(false, a, false, b2, (short)0, acc2, false, false);
    acc3 = __builtin_amdgcn_wmma_f32_16x16x4_f32(false, a, false, b3, (short)0, acc3, false, false);
  }

  // D layout: lane nl holds column n, rows m0 + 8*kh + v (v = VGPR index)
  int mb = m0 + 8 * kh;
  v8f accs[4] = { acc0, acc1, acc2, acc3 };
#pragma unroll
  for (int t = 0; t < 4; t++) {
    int n = t * 16 + nl;
    float bn = bias[n];
    float s = 0.0f, q = 0.0f;
#pragma unroll
    for (int v = 0; v < 8; v++) {
      float d = accs[t][v] + bn;
      h1[(size_t)(mb + v) * HID + n] = d;
      s += d;
      q += d * d;
    }
    // combine the two half-waves (same column n), then one atomic per column
    s += __shfl_xor(s, 16, 32);
    q += __shfl_xor(q, 16, 32);
    if (kh == 0) {
      atomicAdd(&stats[n], s);
      atomicAdd(&stats[HID + n], q);
    }
  }
}

// ---------------------------------------------------------------------------
// BN finalize: scale = gamma * rsqrt(var + eps), shift = beta - mu * scale
// ---------------------------------------------------------------------------
__global__ void gin_bn(const float* __restrict__ stats,
                       const float* __restrict__ gamma,
                       const float* __restrict__ beta,
                       float* __restrict__ sc) {
  int c = threadIdx.x;
  if (c < HID) {
    const float invN = 1.0f / (float)NNODES;
    float mu    = stats[c] * invN;
    float var   = stats[HID + c] * invN - mu * mu;
    float scale = gamma[c] * rsqrtf(var + BN_EPS);
    sc[c]       = scale;
    sc[HID + c] = beta[c] - mu * scale;
  }
}

// ---------------------------------------------------------------------------
// GEMM2: xout = relu( relu(scale*h1+shift) @ W2 + b2 ).  BN+ReLU folded into
// the A-operand load; fp32 WMMA as in GEMM1.
// ---------------------------------------------------------------------------
__global__ void __launch_bounds__(128)
gin_gemm2(const float* __restrict__ h1, const float* __restrict__ sc,
          const float* __restrict__ W, const float* __restrict__ bias,
          float* __restrict__ xout) {
  __shared__ float sWT[HID * WSTRIDE];
  __shared__ float sS[2 * HID];
  load_wt(sWT, W, threadIdx.x);
  if (threadIdx.x < 2 * HID) sS[threadIdx.x] = sc[threadIdx.x];
  __syncthreads();

  int lane = threadIdx.x & 31;
  int tile = blockIdx.x * 4 + (threadIdx.x >> 5);
  if (tile >= NTILES) return;

  int nl = lane & 15;
  int kh = lane >> 4;
  int m0 = tile * 16;
  const float* hrow = h1 + (size_t)(m0 + nl) * HID;
  const float* w0 = sWT + (size_t)nl * WSTRIDE;

  v8f acc0 = {}, acc1 = {}, acc2 = {}, acc3 = {};
  for (int k = 0; k < HID; k += 4) {
    int ka = k + 2 * kh;
    v2f raw   = *(const v2f*)(hrow + ka);
    v2f scale = *(const v2f*)(sS + ka);          // ds_load_b64
    v2f shift = *(const v2f*)(sS + HID + ka);    // ds_load_b64
    v2f a;
    a.x = fmaxf(raw.x * scale.x + shift.x, 0.0f);
    a.y = fmaxf(raw.y * scale.y + shift.y, 0.0f);
    v2f b0 = *(const v2f*)(w0 + ka);
    v2f b1 = *(const v2f*)(w0 + 16 * WSTRIDE + ka);
    v2f b2 = *(const v2f*)(w0 + 32 * WSTRIDE + ka);
    v2f b3 = *(const v2f*)(w0 + 48 * WSTRIDE + ka);
    acc0 = __builtin_amdgcn_wmma_f32_16x16x4_f32(false, a, false, b0, (short)0, acc0, false, false);
    acc1 = __builtin_amdgcn_wmma_f32_16x16x4_f32(false, a, false, b1, (short)0, acc1, false, false);
    acc2 = __builtin_amdgcn_wmma_f32_16x16x4_f32(false, a, false, b2, (short)0, acc2, false, false);
    acc3 = __builtin_amdgcn_wmma_f32_16x16x4_f32(false, a, false, b3, (short)0, acc3, false, false);
  }

  int mb = m0 + 8 * kh;
  v8f accs[4] = { acc0, acc1, acc2, acc3 };
#pragma unroll
  for (int t = 0; t < 4; t++) {
    int n = t * 16 + nl;
    float bn = bias[n];
#pragma unroll
    for (int v = 0; v < 8; v++) {
      xout[(size_t)(mb + v) * HID + n] = fmaxf(accs[t][v] + bn, 0.0f);
    }
  }
}

// ---------------------------------------------------------------------------
// Host launcher.  Inputs: x, edge_index(int64), edge_atten, W1, b1, gamma,
// beta, W2, b2.  Output: float [NNODES, HID].
// ---------------------------------------------------------------------------
extern "C" void kernel_launch(void* const* d_in, const int* in_sizes, int n_in,
                              void* d_out, int out_size, void* d_ws, size_t ws_size,
                              hipStream_t stream) {
  const float*     x     = (const float*)d_in[0];
  const long long* ei    = (const long long*)d_in[1];
  const float*     atten = (const float*)d_in[2];
  const float*     W1    = (const float*)d_in[3];
  const float*     b1    = (const float*)d_in[4];
  const float*     gamma = (const float*)d_in[5];
  const float*     beta  = (const float*)d_in[6];
  const float*     W2    = (const float*)d_in[7];
  const float*     b2    = (const float*)d_in[8];
  float*           out   = (float*)d_out;

  const size_t nodeBytes = (size_t)NNODES * HID * sizeof(float);
  char* w = (char*)d_ws;
  float* agg   = (float*)(w);                  // scatter accumulator
  float* h1    = (float*)(w + nodeBytes);      // post-GEMM1 activations
  float* xb    = (float*)(w + 2 * nodeBytes);  // layer output ping buffer
  float* stats = (float*)(w + 3 * nodeBytes);  // [0:64]=sum [64:128]=sumsq
  float* sc    = stats + 2 * HID;              // [0:64]=scale [64:128]=shift

  dim3 initGrid(NNODES * (HID / 4) / 256);     // 6250
  dim3 edgeGrid(NEDGES * 32 / 256);            // 200000
  dim3 gemmGrid((NTILES + 3) / 4);             // 1563
  dim3 blk256(256), blk128(128), blk64(64);

  const float* xin = x;
  for (int i = 0; i < 3; i++) {
    float* xout = (i == 2) ? out : xb;
    gin_init <<<initGrid, blk256, 0, stream>>>(xin, agg, stats);
    gin_edge <<<edgeGrid, blk256, 0, stream>>>(xin, ei, atten, agg);
    gin_gemm1<<<gemmGrid, blk128, 0, stream>>>(agg, W1 + i * HID * HID,
                                               b1 + i * HID, h1, stats);
    gin_bn   <<<1,        blk64,  0, stream>>>(stats, gamma + i * HID,
                                               beta + i * HID, sc);
    gin_gemm2<<<gemmGrid, blk128, 0, stream>>>(h1, sc, W2 + i * HID * HID,
                                               b2 + i * HID, xout);
    xin = xout;
  }
}